// GCN_UP_12962211299880
// MI455X (gfx1250) — compile-verified
//
#include <hip/hip_runtime.h>

// ---------------------------------------------------------------------------
// GCN + upsample + NN-gather for MI455X (gfx1250, wave32, WMMA fp32 path).
// Dominant cost: 8192x8192x256 Gram matrix (34.4 GFLOP fp32) fused with the
// per-row argmin so the 256MB distance matrix is never written to memory.
// All GEMMs use V_WMMA_F32_16X16X4_F32. Working set (~100MB) fits in the
// 192MB L2, so tiles are fed straight from global (L2-resident) memory.
// ---------------------------------------------------------------------------

typedef __attribute__((ext_vector_type(2))) float v2f;
typedef __attribute__((ext_vector_type(8))) float v8f;

#define N_NODES 4096
#define D_FEAT  256
#define M2      (2 * N_NODES)   // 8192 upsampled rows

__device__ __forceinline__ v8f wmma_f32(v2f a, v2f b, v8f c) {
    // (neg_a, A, neg_b, B, c_mod, C, reuse_a, reuse_b)
    return __builtin_amdgcn_wmma_f32_16x16x4_f32(
        false, a, false, b, (short)0, c, false, false);
}

// ---------------------------------------------------------------------------
// Generic fp32 WMMA GEMM: C[M,N] = A[M,K] @ B[K,N], row-major.
// Block = 128 threads (4 waves); each wave computes a 32x32 patch as a 2x2
// grid of 16x16 accumulator tiles. Grid = (M/64, N/64). M,N,K % 64 == 0.
//
// 16x16x4 fp32 operand layout (wave32):
//   A frag (16x4): lane L -> m = L&15, holds A[m][2h], A[m][2h+1], h = L>>4
//   B frag (4x16): lane L -> n = L&15, holds B[2h][n], B[2h+1][n]
//   C/D (16x16):   VGPR r -> row = r + 8h, col = L&15
// ---------------------------------------------------------------------------
__global__ void gemm_wmma_f32(const float* __restrict__ A,
                              const float* __restrict__ B,
                              float* __restrict__ C,
                              int K, int lda, int ldb, int ldc) {
    const int lane = threadIdx.x & 31;
    const int wave = threadIdx.x >> 5;
    const int h    = lane >> 4;     // 0 or 1: K sub-pair select
    const int ln   = lane & 15;     // m for A-frag, n for B-frag / C cols
    const int m0   = blockIdx.x * 64 + (wave & 1) * 32;
    const int n0   = blockIdx.y * 64 + (wave >> 1) * 32;

    v8f acc[2][2];
#pragma unroll
    for (int mt = 0; mt < 2; ++mt)
#pragma unroll
        for (int nt = 0; nt < 2; ++nt)
            acc[mt][nt] = (v8f){0.f, 0.f, 0.f, 0.f, 0.f, 0.f, 0.f, 0.f};

#pragma unroll 4
    for (int kk = 0; kk < K; kk += 4) {
        v2f a[2], b[2];
#pragma unroll
        for (int mt = 0; mt < 2; ++mt)
            a[mt] = *(const v2f*)(A + (size_t)(m0 + mt * 16 + ln) * lda
                                    + kk + 2 * h);
#pragma unroll
        for (int nt = 0; nt < 2; ++nt) {
            const float* bp = B + (size_t)(kk + 2 * h) * ldb + n0 + nt * 16 + ln;
            b[nt] = (v2f){bp[0], bp[ldb]};
        }
#pragma unroll
        for (int mt = 0; mt < 2; ++mt)
#pragma unroll
            for (int nt = 0; nt < 2; ++nt)
                acc[mt][nt] = wmma_f32(a[mt], b[nt], acc[mt][nt]);
    }

#pragma unroll
    for (int mt = 0; mt < 2; ++mt)
#pragma unroll
        for (int nt = 0; nt < 2; ++nt)
#pragma unroll
            for (int r = 0; r < 8; ++r)
                C[(size_t)(m0 + mt * 16 + r + 8 * h) * ldc
                  + n0 + nt * 16 + ln] = acc[mt][nt][r];
}

// ---------------------------------------------------------------------------
// Linear 2x upsample (align_corners=False) along rows, fused with per-row
// squared-norm reduction. One block (256 threads) per output row.
//   even i=2k:   k==0   ? h[0]   : 0.25*h[k-1] + 0.75*h[k]
//   odd  i=2k+1: k==N-1 ? h[N-1] : 0.75*h[k]   + 0.25*h[k+1]
// ---------------------------------------------------------------------------
__global__ void upsample_sq(const float* __restrict__ H,
                            float* __restrict__ X2,
                            float* __restrict__ sq) {
    __shared__ float red[D_FEAT];
    const int i = blockIdx.x;     // 0..8191
    const int d = threadIdx.x;    // 0..255
    const int k = i >> 1;
    float v;
    if ((i & 1) == 0) {
        v = (k == 0) ? H[d]
                     : 0.25f * H[(size_t)(k - 1) * D_FEAT + d]
                     + 0.75f * H[(size_t)k * D_FEAT + d];
    } else {
        v = (k == N_NODES - 1)
                ? H[(size_t)k * D_FEAT + d]
                : 0.75f * H[(size_t)k * D_FEAT + d]
                + 0.25f * H[(size_t)(k + 1) * D_FEAT + d];
    }
    X2[(size_t)i * D_FEAT + d] = v;
    red[d] = v * v;
    __syncthreads();
#pragma unroll
    for (int s = 128; s > 0; s >>= 1) {
        if (d < s) red[d] += red[d + s];
        __syncthreads();
    }
    if (d == 0) sq[i] = red[0];
}

// ---------------------------------------------------------------------------
// Fused Gram + nearest-neighbor argmin.
//   argmin_j ( sq[i] + sq[j] - 2*g(i,j) ) == argmin_j ( sq[j] - 2*g(i,j) )
// Block = 128 threads (4 waves); each wave owns 32 rows (2 M-tiles) and sweeps
// all 8192 columns in 64-wide chunks (4 N-tiles): 8 WMMAs per K-step against
// 2 A-frag + 4 B-frag loads, so the row-invariant A traffic is amortized 2x
// and WMMA issue density stays high. Running (min,idx) kept per lane, reduced
// across the 16-lane column group at the end. Grid = M2/128 blocks.
// B fragment for the Gram: WMMA needs B[k][n] = x2[j0+n][k], which is a
// contiguous float2 per lane from row j0+n.
// ---------------------------------------------------------------------------
__global__ void gram_argmin(const float* __restrict__ X2,
                            const float* __restrict__ sq,
                            int* __restrict__ nn) {
    const int lane = threadIdx.x & 31;
    const int wave = threadIdx.x >> 5;
    const int h    = lane >> 4;
    const int ln   = lane & 15;
    const int i0   = blockIdx.x * 128 + wave * 32;

    float minv[2][8];
    int   mini[2][8];
#pragma unroll
    for (int mt = 0; mt < 2; ++mt)
#pragma unroll
        for (int r = 0; r < 8; ++r) {
            minv[mt][r] = __builtin_huge_valf();
            mini[mt][r] = 0;
        }

    for (int j0 = 0; j0 < M2; j0 += 64) {
        v8f acc[2][4];
#pragma unroll
        for (int mt = 0; mt < 2; ++mt)
#pragma unroll
            for (int nt = 0; nt < 4; ++nt)
                acc[mt][nt] = (v8f){0.f, 0.f, 0.f, 0.f, 0.f, 0.f, 0.f, 0.f};

#pragma unroll 4
        for (int kk = 0; kk < D_FEAT; kk += 4) {
            v2f a[2], b[4];
#pragma unroll
            for (int mt = 0; mt < 2; ++mt)
                a[mt] = *(const v2f*)(X2 + (size_t)(i0 + mt * 16 + ln) * D_FEAT
                                         + kk + 2 * h);
#pragma unroll
            for (int nt = 0; nt < 4; ++nt)
                b[nt] = *(const v2f*)(X2 + (size_t)(j0 + nt * 16 + ln) * D_FEAT
                                         + kk + 2 * h);
#pragma unroll
            for (int mt = 0; mt < 2; ++mt)
#pragma unroll
                for (int nt = 0; nt < 4; ++nt)
                    acc[mt][nt] = wmma_f32(a[mt], b[nt], acc[mt][nt]);
        }

#pragma unroll
        for (int nt = 0; nt < 4; ++nt) {
            const int   col = j0 + nt * 16 + ln;
            const float sqc = sq[col];
#pragma unroll
            for (int mt = 0; mt < 2; ++mt)
#pragma unroll
                for (int r = 0; r < 8; ++r) {
                    const int   row  = i0 + mt * 16 + r + 8 * h;
                    const float cand = sqc - 2.0f * acc[mt][nt][r];
                    if (col != row &&
                        (cand < minv[mt][r] ||
                         (cand == minv[mt][r] && col < mini[mt][r]))) {
                        minv[mt][r] = cand;
                        mini[mt][r] = col;
                    }
                }
        }
    }

    // Reduce across the 16 lanes holding different columns of the same row.
#pragma unroll
    for (int mt = 0; mt < 2; ++mt)
#pragma unroll
        for (int r = 0; r < 8; ++r) {
            float v  = minv[mt][r];
            int   id = mini[mt][r];
#pragma unroll
            for (int off = 8; off >= 1; off >>= 1) {
                float ov = __shfl_xor(v, off, 16);
                int   oi = __shfl_xor(id, off, 16);
                if (ov < v || (ov == v && oi < id)) { v = ov; id = oi; }
            }
            if (ln == 0)
                nn[i0 + mt * 16 + r + 8 * h] = id;
        }
}

// out[i][:] = support[nn[i]][:]
__global__ void gather_rows(const float* __restrict__ S,
                            const int* __restrict__ nn,
                            float* __restrict__ out) {
    const int i = blockIdx.x;
    const int d = threadIdx.x;
    out[(size_t)i * D_FEAT + d] = S[(size_t)nn[i] * D_FEAT + d];
}

extern "C" void kernel_launch(void* const* d_in, const int* in_sizes, int n_in,
                              void* d_out, int out_size, void* d_ws, size_t ws_size,
                              hipStream_t stream) {
    (void)in_sizes; (void)n_in; (void)out_size; (void)ws_size;
    const float* x   = (const float*)d_in[0];   // [4096, 256]
    const float* adj = (const float*)d_in[1];   // [4096, 4096]
    const float* W1  = (const float*)d_in[2];   // [256, 256]
    const float* W2  = (const float*)d_in[3];   // [256, 256]
    float* out = (float*)d_out;                 // [8192, 256]

    // Scratch layout (~25 MB)
    float* xw = (float*)d_ws;                          // [4096, 256]
    float* H  = xw + (size_t)N_NODES * D_FEAT;         // [4096, 256]
    float* X2 = H  + (size_t)N_NODES * D_FEAT;         // [8192, 256]
    float* sq = X2 + (size_t)M2 * D_FEAT;              // [8192]
    int*   nn = (int*)(sq + M2);                       // [8192]
    float* S  = (float*)(nn + M2);                     // [8192, 256]

    // xw = x @ W1
    gemm_wmma_f32<<<dim3(N_NODES / 64, D_FEAT / 64), 128, 0, stream>>>(
        x, W1, xw, D_FEAT, D_FEAT, D_FEAT, D_FEAT);
    // H = adj @ xw
    gemm_wmma_f32<<<dim3(N_NODES / 64, D_FEAT / 64), 128, 0, stream>>>(
        adj, xw, H, N_NODES, N_NODES, D_FEAT, D_FEAT);
    // X2 = upsample(H), sq[i] = ||X2[i]||^2
    upsample_sq<<<M2, D_FEAT, 0, stream>>>(H, X2, sq);
    // nn[i] = argmin_j!=i d2(i,j)  (fused Gram + argmin, never materialized)
    gram_argmin<<<M2 / 128, 128, 0, stream>>>(X2, sq, nn);
    // S = X2 @ W2
    gemm_wmma_f32<<<dim3(M2 / 64, D_FEAT / 64), 128, 0, stream>>>(
        X2, W2, S, D_FEAT, D_FEAT, D_FEAT, D_FEAT);
    // out = S[nn]
    gather_rows<<<M2, D_FEAT, 0, stream>>>(S, nn, out);
}